// GCN_45578192945656
// MI455X (gfx1250) — compile-verified
//
#include <hip/hip_runtime.h>

typedef __attribute__((ext_vector_type(2))) float v2f;
typedef __attribute__((ext_vector_type(8))) float v8f;

// ---------------------------------------------------------------------------
// Utility kernels
// ---------------------------------------------------------------------------
__global__ void zero_f32(float* __restrict__ p, int n) {
  unsigned i = blockIdx.x * blockDim.x + threadIdx.x;
  const unsigned stride = gridDim.x * blockDim.x;
  for (; i < (unsigned)n; i += stride) p[i] = 0.0f;
}

__global__ void degree_accum(const int* __restrict__ src, const int* __restrict__ dst,
                             float* __restrict__ outdeg, float* __restrict__ indeg, int E) {
  const int e = blockIdx.x * blockDim.x + threadIdx.x;
  if (e < E) {
    atomicAdd(&outdeg[src[e]], 1.0f);
    atomicAdd(&indeg[dst[e]], 1.0f);
  }
}

__global__ void rsqrt_clamp(float* __restrict__ v, int n) {
  const unsigned i = blockIdx.x * blockDim.x + threadIdx.x;
  if (i < (unsigned)n) v[i] = rsqrtf(fmaxf(v[i], 1.0f));
}

__global__ void pad_w(const float* __restrict__ W, float* __restrict__ Wp,
                      int K, int F, int Fp) {
  const int i = blockIdx.x * blockDim.x + threadIdx.x;
  if (i >= K * Fp) return;
  const int k = i / Fp, c = i % Fp;
  Wp[i] = (c < F) ? W[k * F + c] : 0.0f;
}

// ---------------------------------------------------------------------------
// T[m,n] = out_norm[m] * sum_k X[m,k] * W[k,n]     (K fixed = 128)
//
// Persistent-B design: each wave owns one 16-column tile and preloads its
// entire 128x16 B tile into 64 VGPRs (32 x v2f) ONCE.  The block then
// grid-strides over 16-row strips of X; each strip is staged into LDS with
// GLOBAL_LOAD_ASYNC_TO_LDS_B128 (ASYNCcnt DMA), and the fully unrolled K loop
// issues 32 V_WMMA_F32_16X16X4_F32 fed by bank-conflict-free ds_load_b64.
//
// Operand layouts (ISA 7.12.2):
//   A 16x4 f32 : lane<16 -> M=lane, K={k0,k0+1}; lane>=16 -> M=lane-16, K={k0+2,k0+3}
//   B 4x16 f32 : VGPR0 = rows {k0 | k0+2}, VGPR1 = rows {k0+1 | k0+3}
//   C/D        : VGPR v holds M = v + 8*(lane>=16), N = lane&15
// ---------------------------------------------------------------------------
__global__ __launch_bounds__(256) void gemm16_norm(
    const float* __restrict__ X, const float* __restrict__ W,
    const float* __restrict__ out_norm, float* __restrict__ T,
    int Fp, int nStrips) {
  constexpr int K = 128;
  constexpr int LDS_LD = K + 4;                 // 132: 16B-aligned rows, stride%64banks=4
  __shared__ float lx[16 * LDS_LD];

  const int tid  = threadIdx.x;
  const int wave = tid >> 5, lane = tid & 31;
  const int half = lane >> 4, l16 = lane & 15;
  const int nTiles = Fp >> 4;
  const bool active = wave < nTiles;
  const int n0 = (active ? wave : 0) << 4;

  // Preload this wave's B tile (128 x 16) into registers, once.
  v2f breg[K / 4];
  if (active) {
    const float* wcol = W + n0 + l16 + (half * 2) * Fp;
#pragma unroll
    for (int k0 = 0; k0 < K; k0 += 4) {
      v2f b;
      b.x = wcol[k0 * Fp];
      b.y = wcol[(k0 + 1) * Fp];
      breg[k0 >> 2] = b;
    }
  }

  for (int strip = blockIdx.x; strip < nStrips; strip += gridDim.x) {
    const unsigned m0 = (unsigned)strip << 4;

    // Async-DMA the 16x128 A strip into LDS: 512 x b128 transfers.
    {
      const float* gbase = X + m0 * (unsigned)K;
#pragma unroll
      for (int i = tid; i < 512; i += 256) {
        const int r = i >> 5, c4 = (i & 31) << 2;
        const unsigned lds_off = (unsigned)(uintptr_t)(const void*)&lx[r * LDS_LD + c4];
        const float* g = gbase + r * K + c4;
        asm volatile("global_load_async_to_lds_b128 %0, %1, off"
                     :: "v"(lds_off), "v"(g) : "memory");
      }
      asm volatile("s_wait_asynccnt 0x0" ::: "memory");
    }
    __syncthreads();

    if (active) {
      const float* lxrow = &lx[l16 * LDS_LD + half * 2];
      v8f acc = {};
#pragma unroll
      for (int k0 = 0; k0 < K; k0 += 4) {
        v2f a = *(const v2f*)(lxrow + k0);     // 8B-aligned, conflict-free
        acc = __builtin_amdgcn_wmma_f32_16x16x4_f32(
            /*neg_a=*/false, a, /*neg_b=*/false, breg[k0 >> 2],
            /*c_mod=*/(short)0, acc, /*reuse_a=*/false, /*reuse_b=*/false);
      }
#pragma unroll
      for (int v = 0; v < 8; ++v) {
        const unsigned m = m0 + v + half * 8;
        T[m * (unsigned)Fp + n0 + l16] = acc[v] * out_norm[m];
      }
    }
    __syncthreads();                            // LDS reuse across strips
  }
}

// ---------------------------------------------------------------------------
// Wave-per-edge gather (b128) + fp32 global atomic scatter:  agg[dst] += t[src]
// ---------------------------------------------------------------------------
__global__ __launch_bounds__(256) void edge_scatter(
    const int* __restrict__ src, const int* __restrict__ dst,
    const float* __restrict__ t, float* __restrict__ agg, int E, int Fp) {
  const int wid = (int)(((long long)blockIdx.x * blockDim.x + threadIdx.x) >> 5);
  const int lane = threadIdx.x & 31;
  if (wid >= E) return;
  const unsigned sbase = (unsigned)src[wid] * (unsigned)Fp;
  const unsigned dbase = (unsigned)dst[wid] * (unsigned)Fp;
  for (int f = lane * 4; f < Fp; f += 128) {
    const float4 v = *(const float4*)(t + sbase + f);   // global_load_b128
    atomicAdd(&agg[dbase + f + 0], v.x);
    atomicAdd(&agg[dbase + f + 1], v.y);
    atomicAdd(&agg[dbase + f + 2], v.z);
    atomicAdd(&agg[dbase + f + 3], v.w);
  }
}

// ---------------------------------------------------------------------------
// h = relu(agg * in_norm[row] + b[col]),  Fp == 128 (shift/mask indexing)
// ---------------------------------------------------------------------------
__global__ void bias_relu128(const float* __restrict__ agg, const float* __restrict__ in_norm,
                             const float* __restrict__ b, float* __restrict__ h, int total) {
  const unsigned i = blockIdx.x * blockDim.x + threadIdx.x;
  if (i >= (unsigned)total) return;
  const float v = fmaf(agg[i], in_norm[i >> 7], b[i & 127]);
  h[i] = fmaxf(v, 0.0f);
}

// ---------------------------------------------------------------------------
// Wave-per-node: y = x*in_norm + b;  out = y - max(y) - log(sum(exp(y-max)))
// ---------------------------------------------------------------------------
__global__ __launch_bounds__(256) void norm_bias_logsoftmax(
    const float* __restrict__ agg, const float* __restrict__ in_norm,
    const float* __restrict__ b, float* __restrict__ out, int N, int Fp, int F) {
  const int node = (int)(((long long)blockIdx.x * blockDim.x + threadIdx.x) >> 5);
  const int lane = threadIdx.x & 31;
  if (node >= N) return;
  const float inn = in_norm[node];
  const unsigned base = (unsigned)node * (unsigned)Fp;
  const int c0 = lane, c1 = lane + 32;
  const float NEG_INF = -__builtin_inff();
  float x0 = (c0 < F) ? fmaf(agg[base + c0], inn, b[c0]) : NEG_INF;
  float x1 = (c1 < F) ? fmaf(agg[base + c1], inn, b[c1]) : NEG_INF;
  float m = fmaxf(x0, x1);
#pragma unroll
  for (int off = 16; off > 0; off >>= 1) m = fmaxf(m, __shfl_xor(m, off, 32));
  float s = ((c0 < F) ? __expf(x0 - m) : 0.0f) + ((c1 < F) ? __expf(x1 - m) : 0.0f);
#pragma unroll
  for (int off = 16; off > 0; off >>= 1) s += __shfl_xor(s, off, 32);
  const float ls = __logf(s);
  if (c0 < F) out[(unsigned)node * (unsigned)F + c0] = x0 - m - ls;
  if (c1 < F) out[(unsigned)node * (unsigned)F + c1] = x1 - m - ls;
}

// ---------------------------------------------------------------------------
extern "C" void kernel_launch(void* const* d_in, const int* in_sizes, int n_in,
                              void* d_out, int out_size, void* d_ws, size_t ws_size,
                              hipStream_t stream) {
  const float* x  = (const float*)d_in[0];
  const int*   ei = (const int*)d_in[1];
  const float* W1 = (const float*)d_in[2];
  const float* b1 = (const float*)d_in[3];
  const float* W2 = (const float*)d_in[4];
  const float* b2 = (const float*)d_in[5];
  const float* W3 = (const float*)d_in[6];
  const float* b3 = (const float*)d_in[7];
  float* out = (float*)d_out;

  const int N = in_sizes[0] / 128;   // 100000 (multiple of 16)
  const int E = in_sizes[1] / 2;     // 1600000
  const int K = 128, H = 128, FO = 40, FOP = 48;
  const int* src = ei;
  const int* dst = ei + E;

  // Workspace (floats): out_norm[N] | in_norm[N] | bufA[N*H] | bufB[N*H] | W3p[K*FOP]
  float* ws   = (float*)d_ws;
  float* outn = ws;
  float* inn  = outn + N;
  float* bufA = inn + N;
  float* bufB = bufA + (size_t)N * H;
  float* w3p  = bufB + (size_t)N * H;

  const int T256 = 256;
  auto cdiv = [](long long a, long long bb) { return (int)((a + bb - 1) / bb); };
  const int nStrips = N / 16;                 // 6250
  const int NH = N * H;                       // 12.8M, fits int
  const int gemmBlocks = 1024;

  // Degrees -> symmetric norms
  zero_f32<<<1024, T256, 0, stream>>>(outn, 2 * N);
  degree_accum<<<cdiv(E, T256), T256, 0, stream>>>(src, dst, outn, inn, E);
  rsqrt_clamp<<<cdiv(2 * N, T256), T256, 0, stream>>>(outn, 2 * N);

  // Layer 1: t1 = (x@W1)*outn ; scatter ; h1 = relu(agg*inn + b1)
  gemm16_norm<<<gemmBlocks, T256, 0, stream>>>(x, W1, outn, bufA, H, nStrips);
  zero_f32<<<2048, T256, 0, stream>>>(bufB, NH);
  edge_scatter<<<cdiv((long long)E * 32, T256), T256, 0, stream>>>(src, dst, bufA, bufB, E, H);
  bias_relu128<<<cdiv(NH, T256), T256, 0, stream>>>(bufB, inn, b1, bufA, NH);

  // Layer 2
  gemm16_norm<<<gemmBlocks, T256, 0, stream>>>(bufA, W2, outn, bufB, H, nStrips);
  zero_f32<<<2048, T256, 0, stream>>>(bufA, NH);
  edge_scatter<<<cdiv((long long)E * 32, T256), T256, 0, stream>>>(src, dst, bufB, bufA, E, H);
  bias_relu128<<<cdiv(NH, T256), T256, 0, stream>>>(bufA, inn, b2, bufB, NH);

  // Layer 3 (40 cols padded to 48 so WMMA tiles are full; pad cols stay 0)
  pad_w<<<cdiv(K * FOP, T256), T256, 0, stream>>>(W3, w3p, K, FO, FOP);
  gemm16_norm<<<gemmBlocks, T256, 0, stream>>>(bufB, w3p, outn, bufA, FOP, nStrips);
  zero_f32<<<1024, T256, 0, stream>>>(bufB, N * FOP);
  edge_scatter<<<cdiv((long long)E * 32, T256), T256, 0, stream>>>(src, dst, bufA, bufB, E, FOP);
  norm_bias_logsoftmax<<<cdiv((long long)N * 32, T256), T256, 0, stream>>>(
      bufB, inn, b3, out, N, FOP, FO);
}